// FPSSampler_5342939316675
// MI455X (gfx1250) — compile-verified
//
#include <hip/hip_runtime.h>
#include <stdint.h>

// FPS (farthest point sampling): x (B=64, N=65536, 3) f32 -> y (B, 2048, 3) f32.
// One workgroup per batch; 1024 threads = 32 wave32 waves; 64 points/thread.
// 24576 points cached in LDS (288 KB): first 16384 staged by the Tensor Data
// Mover (one tensor_load_to_lds), remaining 8192 by per-lane async global->LDS
// copies. The rest of the batch streams from L2 (48 MB dataset is L2-resident).

#define N_PTS   65536
#define M_OUT   2048
#define THREADS 1024
#define WAVES   (THREADS / 32)
#define LDS_PTS 24576                  // points resident in LDS (AoS, 12 B each)
#define K_LDS   (LDS_PTS / THREADS)    // 24 LDS-resident points per thread
#define K_TOT   (N_PTS / THREADS)      // 64 points per thread total

#define ROW_DW    3072u                // one "row" = 1024 points * 3 dwords
#define TDM_ROWS  16u                  // rows staged by TDM  -> points 0..16383
#define ASYNC_PT0 (1024 * (int)TDM_ROWS)        // first async-staged point
#define ASYNC_B0  (ASYNC_PT0 * 12)              // byte offset of async region
#define ASYNC_BYTES ((LDS_PTS - ASYNC_PT0) * 12)

typedef unsigned uint32x4 __attribute__((ext_vector_type(4)));
typedef unsigned uint32x8 __attribute__((ext_vector_type(8)));

__launch_bounds__(THREADS, 1)
__global__ void fps_kernel(const float* __restrict__ x, float* __restrict__ y) {
  __shared__ __align__(16) float s_pts[LDS_PTS * 3];   // 294912 B
  __shared__ float    s_red[WAVES];
  __shared__ float    s_gmax;
  __shared__ unsigned s_idx;

  const int b = blockIdx.x;
  const float* __restrict__ xb = x + (size_t)b * (size_t)(N_PTS * 3);
  float* __restrict__ yb = y + (size_t)b * (size_t)(M_OUT * 3);

  const int tid  = (int)threadIdx.x;
  const int lane = tid & 31;
  const int wid  = tid >> 5;

  const unsigned lbase = (unsigned)(uintptr_t)&s_pts[0];   // low 32 bits = LDS offset

  // ---- Stage points 0..16383 with the Tensor Data Mover (one DMA) ----
  if (wid == 0) {
    const uint64_t ga = (uint64_t)(uintptr_t)xb;
    uint32x4 g0;
    g0.x = 1u;                                        // count=1, user descriptor
    g0.y = lbase;                                     // lds_addr (bytes)
    g0.z = (unsigned)(ga & 0xffffffffu);              // global_addr[31:0]
    g0.w = (unsigned)((ga >> 32) & 0x01ffffffu)       // global_addr[56:32]
         | 0x80000000u;                               // type=2 ("image")
    uint32x8 g1;
    g1[0] = (2u << 16);                               // data_size=2 (4 bytes)
    g1[1] = (ROW_DW & 0xffffu) << 16;                 // tensor_dim0 lo16 (=3072)
    g1[2] = ((ROW_DW >> 16) & 0xffffu)                // tensor_dim0 hi16
          | (64u << 16);                              // tensor_dim1 lo16 (=64 rows)
    g1[3] = (ROW_DW << 16);                           // tensor_dim1 hi16=0 | tile_dim0=3072
    g1[4] = TDM_ROWS;                                 // tile_dim1=16, tile_dim2=0
    g1[5] = ROW_DW;                                   // tensor_dim0_stride lo32
    g1[6] = 0u;                                       // stride0 hi16 | stride1 lo16
    g1[7] = 0u;                                       // stride1 hi32
    asm volatile("tensor_load_to_lds %0, %1" :: "s"(g0), "s"(g1) : "memory");
  }

  // ---- Stage points 16384..24575 with per-lane async global->LDS copies ----
  {
    const char* gbase = (const char*)xb;
    #pragma unroll
    for (int c = 0; c < ASYNC_BYTES / (THREADS * 16); ++c) {
      const unsigned off = (unsigned)(ASYNC_B0 + tid * 16 + c * (THREADS * 16));
      const unsigned laddr = lbase + off;
      const char* gaddr = gbase + off;
      asm volatile("global_load_async_to_lds_b128 %0, %1, off"
                   :: "v"(laddr), "v"(gaddr) : "memory");
    }
    asm volatile("s_wait_asynccnt 0" ::: "memory");
    __builtin_amdgcn_s_wait_tensorcnt(0);   // only wave 0 has TENSORcnt>0; harmless elsewhere
    __syncthreads();
  }

  // Running min-distance per owned point, kept in VGPRs.
  float mind[K_TOT];
  #pragma unroll
  for (int k = 0; k < K_TOT; ++k) mind[k] = __builtin_inff();

  // First selected index is 0 (matches reference scan init).
  float p0 = xb[0], p1 = xb[1], p2 = xb[2];

  for (int it = 0; it < M_OUT; ++it) {
    // Emit the currently selected point (reference records `last` each step).
    if (tid == 0) {
      yb[3 * it + 0] = p0;
      yb[3 * it + 1] = p1;
      yb[3 * it + 2] = p2;
    }

    float    bmax = -1.0f;
    unsigned bidx = 0u;

    // LDS-resident points (ds_load path).
    #pragma unroll
    for (int k = 0; k < K_LDS; ++k) {
      const int i = tid + k * THREADS;
      const float qx = s_pts[3 * i + 0];
      const float qy = s_pts[3 * i + 1];
      const float qz = s_pts[3 * i + 2];
      const float dx = qx - p0, dy = qy - p1, dz = qz - p2;
      const float d  = __builtin_fmaf(dz, dz, __builtin_fmaf(dy, dy, dx * dx));
      const float nm = fminf(mind[k], d);
      mind[k] = nm;
      if (nm > bmax) { bmax = nm; bidx = (unsigned)i; }  // strict > keeps first occurrence
    }
    // L2-resident points (vmem path; addresses identical every iteration).
    #pragma unroll
    for (int k = K_LDS; k < K_TOT; ++k) {
      const int i = tid + k * THREADS;
      const float* q = xb + 3 * (size_t)i;
      const float dx = q[0] - p0, dy = q[1] - p1, dz = q[2] - p2;
      const float d  = __builtin_fmaf(dz, dz, __builtin_fmaf(dy, dy, dx * dx));
      const float nm = fminf(mind[k], d);
      mind[k] = nm;
      if (nm > bmax) { bmax = nm; bidx = (unsigned)i; }
    }

    // Wave32 max reduction.
    float v = bmax;
    #pragma unroll
    for (int off = 16; off > 0; off >>= 1)
      v = fmaxf(v, __shfl_xor(v, off, 32));
    if (lane == 0) s_red[wid] = v;
    __syncthreads();

    // Cross-wave reduction by wave 0 (32 partials == 32 lanes).
    if (wid == 0) {
      float w = s_red[lane];
      #pragma unroll
      for (int off = 16; off > 0; off >>= 1)
        w = fmaxf(w, __shfl_xor(w, off, 32));
      if (lane == 0) { s_gmax = w; s_idx = 0xFFFFFFFFu; }
    }
    __syncthreads();

    // First-occurrence argmax: fmax chains return one of their (bit-exact)
    // inputs, so equality against the broadcast max is exact. ds_min_u32
    // picks the smallest global index among ties -> matches jnp.argmax.
    const float gmax = s_gmax;
    if (bmax == gmax) atomicMin(&s_idx, bidx);
    __syncthreads();

    unsigned widx = s_idx;
    widx = (unsigned)__builtin_amdgcn_readfirstlane((int)widx);  // uniform -> scalar loads
    p0 = xb[3 * (size_t)widx + 0];
    p1 = xb[3 * (size_t)widx + 1];
    p2 = xb[3 * (size_t)widx + 2];
  }
}

extern "C" void kernel_launch(void* const* d_in, const int* in_sizes, int n_in,
                              void* d_out, int out_size, void* d_ws, size_t ws_size,
                              hipStream_t stream) {
  (void)n_in; (void)out_size; (void)d_ws; (void)ws_size;
  const float* x = (const float*)d_in[0];
  float* y = (float*)d_out;
  const int B = in_sizes[0] / (N_PTS * 3);   // 64 for the reference shapes
  fps_kernel<<<dim3(B), dim3(THREADS), 0, stream>>>(x, y);
}